// LocalMHSA_34754875359462
// MI455X (gfx1250) — compile-verified
//
#include <hip/hip_runtime.h>
#include <cstdint>

// ---------------------------------------------------------------------------
// LocalMHSA for MI455X (gfx1250, wave32).
//   Stage 1: qkv = x @ W_qkv^T + b_qkv          (fp32 WMMA 16x16x4)
//   Stage 2: neighborhood attention (gather + softmax over K=32), L2-resident
//   Stage 3: out = attn @ W_proj^T + b_proj     (fp32 WMMA 16x16x4)
// ---------------------------------------------------------------------------

typedef float v2f __attribute__((ext_vector_type(2)));
typedef float v8f __attribute__((ext_vector_type(8)));

#define TB 256
#define WAVES_PER_BLOCK 8

#define BATCH  2
#define NTOK   8192
#define KNBR   32
#define NHEAD  8
#define DHEAD  32
#define DMODEL 256
#define QKVW   768
#define ATTN_SCALE 0.17677669529663687f   // 1/sqrt(32)

// ---------------------------------------------------------------------------
// C[m][n] = sum_k A[m][k] * W[n][k] + bias[n]
// One wave computes one 16x16 output tile via V_WMMA_F32_16X16X4_F32.
// A-frag: lane<16 -> (M=lane, K=k0..k0+1), lane>=16 -> (M=lane-16, K=k0+2..3)
// B-frag: same pattern with N instead of M; B[k][n] = W[n][k] so both frags
// are contiguous float2 loads along the K axis of their respective rows.
// ---------------------------------------------------------------------------
__global__ __launch_bounds__(TB) void gemm_xwT_bias_wmma(
    const float* __restrict__ A, const float* __restrict__ W,
    const float* __restrict__ bias, float* __restrict__ C,
    int M, int Nout, int Kd)
{
  const int waveId = blockIdx.x * WAVES_PER_BLOCK + (threadIdx.x >> 5);
  const int tilesN = Nout >> 4;
  const int tm = waveId / tilesN;            // 16-row block
  const int tn = waveId % tilesN;            // 16-col block (waves in a block
                                             //  share A rows -> L0 reuse)
  if ((tm << 4) >= M) return;                // uniform per wave

  const int lane = threadIdx.x & 31;
  const int half = lane >> 4;                // 0: K pair {0,1}; 1: {2,3}
  const int l15  = lane & 15;

  const float* __restrict__ arow = A + (size_t)(tm * 16 + l15) * Kd + 2 * half;
  const float* __restrict__ wrow = W + (size_t)(tn * 16 + l15) * Kd + 2 * half;

  v8f acc = {0.f, 0.f, 0.f, 0.f, 0.f, 0.f, 0.f, 0.f};
#pragma unroll 8
  for (int k0 = 0; k0 < Kd; k0 += 4) {
    v2f a = *reinterpret_cast<const v2f*>(arow + k0);
    v2f b = *reinterpret_cast<const v2f*>(wrow + k0);
    // (neg_a, A, neg_b, B, c_mod, C, reuse_a, reuse_b)
    acc = __builtin_amdgcn_wmma_f32_16x16x4_f32(
        false, a, false, b, (short)0, acc, false, false);
  }

  // D layout: VGPR r, lane L -> row = r + 8*(L>=16), col = L&15
  const float bv   = bias[tn * 16 + l15];
  const int rbase  = tm * 16 + 8 * half;
  const int col    = tn * 16 + l15;
#pragma unroll
  for (int r = 0; r < 8; ++r)
    C[(size_t)(rbase + r) * Nout + col] = acc[r] + bv;
}

// ---------------------------------------------------------------------------
// One block per token (8 waves = 8 heads). qkv row layout: [q|k|v], head h at
// offset h*32 within each 256-wide segment. Scores: lane k owns neighbor k
// (8x float4 K-row loads). Softmax: wave32 shfl_xor butterflies. Output:
// lane d owns dim d, weights broadcast with __shfl -> no reductions, fully
// coalesced 128B V gathers (served by the 192MB L2 holding qkv).
// ---------------------------------------------------------------------------
__global__ __launch_bounds__(TB) void local_attn_kernel(
    const float* __restrict__ qkv, const long long* __restrict__ nbr_idx,
    const unsigned char* __restrict__ nbr_mask, float* __restrict__ attn_out)
{
  const int n = blockIdx.x;
  const int b = blockIdx.y;
  const size_t row = (size_t)b * NTOK + n;

  __shared__ long long s_idx[KNBR];
  __shared__ float     s_q[DMODEL];

  const int tid = threadIdx.x;
  if (tid < KNBR) s_idx[tid] = nbr_idx[row * KNBR + tid];
  s_q[tid] = qkv[row * QKVW + tid];
  __syncthreads();

  const int h    = tid >> 5;
  const int lane = tid & 31;

  // ---- scores: this lane's neighbor ----
  const size_t krow =
      ((size_t)b * NTOK + (size_t)s_idx[lane]) * QKVW + DMODEL + h * DHEAD;
  const float4* kp = reinterpret_cast<const float4*>(qkv + krow);
  float sc = 0.f;
#pragma unroll
  for (int d4 = 0; d4 < DHEAD / 4; ++d4) {
    float4 kv = kp[d4];
    sc += s_q[h * DHEAD + 4 * d4 + 0] * kv.x;
    sc += s_q[h * DHEAD + 4 * d4 + 1] * kv.y;
    sc += s_q[h * DHEAD + 4 * d4 + 2] * kv.z;
    sc += s_q[h * DHEAD + 4 * d4 + 3] * kv.w;
  }
  sc *= ATTN_SCALE;
  if (!nbr_mask[row * KNBR + lane]) sc = -__builtin_huge_valf();

  // ---- softmax over K=32 (one wave) ----
  float m = sc;
#pragma unroll
  for (int off = 16; off; off >>= 1) m = fmaxf(m, __shfl_xor(m, off, 32));
  float e = __expf(sc - m);
  float s = e;
#pragma unroll
  for (int off = 16; off; off >>= 1) s += __shfl_xor(s, off, 32);
  const float w = e / s;

  // ---- weighted V sum: lane owns output dim `lane` of head h ----
  float acc = 0.f;
#pragma unroll 4
  for (int k = 0; k < KNBR; ++k) {
    const float wk = __shfl(w, k, 32);
    const size_t vrow =
        ((size_t)b * NTOK + (size_t)s_idx[k]) * QKVW + 2 * DMODEL + h * DHEAD;
    acc += wk * qkv[vrow + lane];
  }
  attn_out[row * DMODEL + h * DHEAD + lane] = acc;
}

// ---------------------------------------------------------------------------
extern "C" void kernel_launch(void* const* d_in, const int* in_sizes, int n_in,
                              void* d_out, int out_size, void* d_ws, size_t ws_size,
                              hipStream_t stream) {
  const float*         x      = (const float*)d_in[0];
  const long long*     idx    = (const long long*)d_in[1];
  const unsigned char* mask   = (const unsigned char*)d_in[2];
  const float*         W_qkv  = (const float*)d_in[3];
  const float*         b_qkv  = (const float*)d_in[4];
  const float*         W_proj = (const float*)d_in[5];
  const float*         b_proj = (const float*)d_in[6];
  float*               out    = (float*)d_out;

  const int M = BATCH * NTOK;                       // 16384 rows

  // workspace: qkv (M x 768 f32, 50.3MB) + attn_out (M x 256 f32, 16.8MB);
  // both fully written before read every call -> deterministic.
  float* qkv      = (float*)d_ws;
  float* attn_out = qkv + (size_t)M * QKVW;

  {   // Stage 1: QKV projection, 1024 x 48 tiles
    const int tiles = (M / 16) * (QKVW / 16);
    gemm_xwT_bias_wmma<<<tiles / WAVES_PER_BLOCK, TB, 0, stream>>>(
        x, W_qkv, b_qkv, qkv, M, QKVW, DMODEL);
  }
  {   // Stage 2: neighborhood attention, one block per token
    dim3 grid(NTOK, BATCH);
    local_attn_kernel<<<grid, TB, 0, stream>>>(qkv, idx, mask, attn_out);
  }
  {   // Stage 3: output projection, 1024 x 16 tiles
    const int tiles = (M / 16) * (DMODEL / 16);
    gemm_xwT_bias_wmma<<<tiles / WAVES_PER_BLOCK, TB, 0, stream>>>(
        attn_out, W_proj, b_proj, out, M, DMODEL, DMODEL);
  }
}